// MolecularCliffordEPModel_48627619725328
// MI455X (gfx1250) — compile-verified
//
#include <hip/hip_runtime.h>
#include <hip/hip_bf16.h>

typedef __attribute__((ext_vector_type(2))) float v2f;
typedef __attribute__((ext_vector_type(8))) float v8f;

// Reorder sign of e_a * e_b for Euclidean Cl(3,0): (-1)^(#transpositions).
// C[a,b,a^b] = blade_sign(a,b) (metric all +1).
__device__ __forceinline__ float blade_sign(int a, int b) {
    int swaps = 0;
    int t = a >> 1;
    while (t) { swaps += __popc(t & b); t >>= 1; }
    return (swaps & 1) ? -1.0f : 1.0f;
}

// ---------------------------------------------------------------------------
// Stage 1: A[o,p] = c * sum_{j,k : j^k = p} C[p,j,k] * s[k] * M[o,j,k]
//          M[o,j,k] = sum_h W_in[h,j] * W_out[o,h,k]
// One block per o (O=128), 64 threads = one (j,k) pair each.
// ---------------------------------------------------------------------------
__global__ void build_A_kernel(const float* __restrict__ W_in,   // [H, 8]
                               const float* __restrict__ W_out,  // [O, H, 8]
                               float* __restrict__ Amat,         // [O, 8]
                               int H)
{
    const int o = blockIdx.x;
    const int t = threadIdx.x;   // 0..63
    const int j = t >> 3;
    const int k = t & 7;

    const float* wo = W_out + (size_t)o * H * 8;
    float acc = 0.0f;
    #pragma unroll 4
    for (int h = 0; h < H; ++h) {
        acc += W_in[h * 8 + j] * wo[h * 8 + k];
    }

    // c = 1 - (1-dt)^10 : closed form of 10 gradient-descent steps from h0=0
    float decay = 1.0f;
    #pragma unroll
    for (int n = 0; n < 10; ++n) decay *= (1.0f - 0.1f);
    const float cc = 1.0f - decay;

    const int p = j ^ k;                                  // x blade index fed by (j,k)
    const float w = blade_sign(p, j) * blade_sign(k, k) * cc;

    __shared__ float buf[64];
    buf[t] = w * acc;
    __syncthreads();

    if (t < 8) {
        float s = 0.0f;
        #pragma unroll
        for (int jj = 0; jj < 8; ++jj) s += buf[jj * 8 + (jj ^ t)];
        Amat[o * 8 + t] = s;
    }
}

// ---------------------------------------------------------------------------
// Stage 2: out[B,O] = X[B,8] @ A^T[8,O] via V_WMMA_F32_16X16X4_F32.
// One wave32 per 16x16 output tile; K=8 -> two K=4 WMMA accumulations.
// Operand layouts per CDNA5 ISA 7.12.2 (32-bit A 16x4 / C,D 16x16).
// ---------------------------------------------------------------------------
__global__ __launch_bounds__(256) void gemm_out_wmma(
    const float* __restrict__ X,     // [B, 8]
    const float* __restrict__ Amat,  // [O, 8]
    float* __restrict__ out,         // [B, O]
    int O, int nTiles)               // nTiles = (B/16)*(O/16)
{
    const int lane = threadIdx.x & 31;
    const int wave = threadIdx.x >> 5;
    const int tile = blockIdx.x * 8 + wave;
    if (tile >= nTiles) return;      // whole-wave early out: EXEC stays all-ones

    const int tilesO = O >> 4;
    const int b0 = (tile / tilesO) * 16;
    const int o0 = (tile % tilesO) * 16;

    const int half = lane >> 4;      // 0: lanes 0-15, 1: lanes 16-31
    const int idx  = lane & 15;      // M (for A-operand) / N (for B-operand)

    v8f acc = {};

    #pragma unroll
    for (int k0 = 0; k0 < 8; k0 += 4) {
        // A-operand 16x4: VGPR0 = K=k0 (lanes 0-15) | K=k0+2 (lanes 16-31),
        //                 VGPR1 = K=k0+1            | K=k0+3
        v2f a, bm;
        const float* xr = X + (size_t)(b0 + idx) * 8 + k0 + 2 * half;
        a.x = xr[0];
        a.y = xr[1];
        // B-operand 4x16 (mirror striping): B[k][n] = Amat[o0+n][k]
        const float* ar = Amat + (size_t)(o0 + idx) * 8 + k0 + 2 * half;
        bm.x = ar[0];
        bm.y = ar[1];
        // (neg_a, A, neg_b, B, c_mod, C, reuse_a, reuse_b)
        acc = __builtin_amdgcn_wmma_f32_16x16x4_f32(
            false, a, false, bm, (short)0, acc, false, false);
    }

    // C/D 16x16 f32: VGPR v holds row M = v + 8*half, col N = idx
    float* orow = out + (size_t)(b0 + 8 * half) * O + o0 + idx;
    #pragma unroll
    for (int v = 0; v < 8; ++v) {
        orow[(size_t)v * O] = acc[v];
    }
}

extern "C" void kernel_launch(void* const* d_in, const int* in_sizes, int n_in,
                              void* d_out, int out_size, void* d_ws, size_t ws_size,
                              hipStream_t stream) {
    const float* x_mv  = (const float*)d_in[0];   // [B, 8]
    const float* W_in  = (const float*)d_in[1];   // [H, 1, 8]
    const float* W_out = (const float*)d_in[2];   // [O, H, 8]
    float* out = (float*)d_out;                   // [B, O]
    float* Amat = (float*)d_ws;                   // [O, 8] scratch (4 KB)

    const int B = in_sizes[0] / 8;                // 4096
    const int H = in_sizes[1] / 8;                // 512
    const int O = in_sizes[2] / (H * 8);          // 128

    build_A_kernel<<<O, 64, 0, stream>>>(W_in, W_out, Amat, H);

    const int nTiles = (B / 16) * (O / 16);       // 2048 tiles, 1 wave each
    const int blocks = (nTiles + 7) / 8;          // 8 waves (256 threads) / block
    gemm_out_wmma<<<blocks, 256, 0, stream>>>(x_mv, Amat, out, O, nTiles);
}